// OrthogonalMatrixLoss_37520834298330
// MI455X (gfx1250) — compile-verified
//
#include <hip/hip_runtime.h>

typedef float v2f __attribute__((ext_vector_type(2)));
typedef float v8f __attribute__((ext_vector_type(8)));

#define BATCH 64
#define DIM 1024
#define ROWSPLIT 4
#define KCHUNKS 8                      // 1024 cols / (8 waves * 16 cols)
#define NBLOCKS1 (BATCH * KCHUNKS * ROWSPLIT)
#define ROWS_PER_BLOCK (DIM / ROWSPLIT)  // 256

// Pass 1: per-(batch, column, row-chunk) column sums via WMMA against an
// all-ones A matrix, plus per-block sum-of-squares on the VALU.
__global__ __launch_bounds__(256)
void OrthLoss_colsum_ssq(const float* __restrict__ x,
                         float* __restrict__ wsColsum,
                         float* __restrict__ wsSsq) {
    const int bid = blockIdx.x;
    const int r   = bid % ROWSPLIT;
    const int kc  = (bid / ROWSPLIT) % KCHUNKS;
    const int b   = bid / (ROWSPLIT * KCHUNKS);

    const int tid    = threadIdx.x;
    const int wave   = tid >> 5;
    const int lane   = tid & 31;
    const int col    = kc * 128 + wave * 16 + (lane & 15);
    const int rowoff = lane >> 4;      // 0 or 1: which of the two rows this half-wave holds

    const float* __restrict__ base = x + (size_t)b * DIM * DIM;

    // A = 16x4 all-ones -> D[m,n] = sum over the 4 K-rows of B column n.
    // Since A is all-ones, any unique row->K assignment in B is valid, which
    // lets us use a fully-coalesced load pattern for B.
    v2f ones; ones[0] = 1.0f; ones[1] = 1.0f;
    v8f acc0 = {};
    v8f acc1 = {};
    float ssq = 0.0f;

    const int i0 = r * ROWS_PER_BLOCK;
    #pragma unroll 4
    for (int i = i0; i < i0 + ROWS_PER_BLOCK; i += 8) {
        const size_t rbase = (size_t)i * DIM + (size_t)col;
        float e0 = base[rbase + (size_t)(rowoff    ) * DIM];
        float e1 = base[rbase + (size_t)(rowoff + 2) * DIM];
        float e2 = base[rbase + (size_t)(rowoff + 4) * DIM];
        float e3 = base[rbase + (size_t)(rowoff + 6) * DIM];
        ssq = fmaf(e0, e0, ssq);
        ssq = fmaf(e1, e1, ssq);
        ssq = fmaf(e2, e2, ssq);
        ssq = fmaf(e3, e3, ssq);
        v2f b0; b0[0] = e0; b0[1] = e1;   // rows i..i+3  (4x16 tile)
        v2f b1; b1[0] = e2; b1[1] = e3;   // rows i+4..i+7
        acc0 = __builtin_amdgcn_wmma_f32_16x16x4_f32(
                   false, ones, false, b0, (short)0, acc0, false, false);
        acc1 = __builtin_amdgcn_wmma_f32_16x16x4_f32(
                   false, ones, false, b1, (short)0, acc1, false, false);
    }

    // Every M-row of D is identical; element 0 of the accumulator in lane L
    // holds the column sum for column k0 + (L % 16). Lanes 16..31 duplicate.
    if (lane < 16) {
        wsColsum[((size_t)b * DIM + (size_t)col) * ROWSPLIT + r] = acc0[0] + acc1[0];
    }

    // Block-reduce the sum of squares.
    __shared__ float red[256];
    red[tid] = ssq;
    __syncthreads();
    for (int s = 128; s > 0; s >>= 1) {
        if (tid < s) red[tid] += red[tid + s];
        __syncthreads();
    }
    if (tid == 0) wsSsq[bid] = red[0];
}

// Pass 2: single block, deterministic combine:
// out = (sum_{b,k} colsum^2 - sum x^2) / BATCH
__global__ __launch_bounds__(1024)
void OrthLoss_finalize(const float* __restrict__ wsColsum,
                       const float* __restrict__ wsSsq,
                       float* __restrict__ out) {
    const int tid = threadIdx.x;
    float acc = 0.0f;
    for (int idx = tid; idx < BATCH * DIM; idx += 1024) {
        const float* p = wsColsum + (size_t)idx * ROWSPLIT;
        float s = (p[0] + p[1]) + (p[2] + p[3]);
        acc = fmaf(s, s, acc);
    }
    for (int j = tid; j < NBLOCKS1; j += 1024) {
        acc -= wsSsq[j];
    }
    __shared__ float red[1024];
    red[tid] = acc;
    __syncthreads();
    for (int s = 512; s > 0; s >>= 1) {
        if (tid < s) red[tid] += red[tid + s];
        __syncthreads();
    }
    if (tid == 0) out[0] = red[0] * (1.0f / (float)BATCH);
}

extern "C" void kernel_launch(void* const* d_in, const int* in_sizes, int n_in,
                              void* d_out, int out_size, void* d_ws, size_t ws_size,
                              hipStream_t stream) {
    (void)in_sizes; (void)n_in; (void)out_size; (void)ws_size;
    const float* x = (const float*)d_in[0];

    float* wsColsum = (float*)d_ws;                                   // 64*1024*4 floats = 1 MB
    float* wsSsq    = wsColsum + (size_t)BATCH * DIM * ROWSPLIT;      // 2048 floats

    OrthLoss_colsum_ssq<<<NBLOCKS1, 256, 0, stream>>>(x, wsColsum, wsSsq);
    OrthLoss_finalize<<<1, 1024, 0, stream>>>(wsColsum, wsSsq, (float*)d_out);
}